// GCNDomainAdaptation_46471546143560
// MI455X (gfx1250) — compile-verified
//
#include <hip/hip_runtime.h>
#include <hip/hip_bf16.h>
#include <math.h>

typedef float v2f __attribute__((ext_vector_type(2)));
typedef float v8f __attribute__((ext_vector_type(8)));

#define H_DIM 128
#define C_DIM 8
// Wp2: [H_DIM/2][16] float2 pairs (W[kk][m], W[kk+1][m]), cols 8..15 zero-padded
#define WP2_ELEMS (H_DIM / 2 * 16)

// ---------------- K0: init scratch (out=0, deg=1.0 self-loop, acc=0) ----------------
__global__ void k_init(float* __restrict__ out, float* __restrict__ deg,
                       float* __restrict__ acc, int N) {
  int i = blockIdx.x * blockDim.x + threadIdx.x;
  int total = N * C_DIM;
  if (i < total) out[i] = 0.0f;
  if (i < N) deg[i] = 1.0f;  // self loop contributes 1 to every node's degree
  if (i == 0) acc[0] = 0.0f;
}

// ---------------- K0b: build padded/interleaved B-matrix ----------------
// Wp2[kk2*16 + m] = (W[2*kk2][m], W[2*kk2+1][m]) for m<8, else (0,0)
__global__ void k_prepW(const float* __restrict__ W, float* __restrict__ Wp2) {
  int i = blockIdx.x * blockDim.x + threadIdx.x;  // 0..WP2_ELEMS-1
  if (i >= WP2_ELEMS) return;
  int kk2 = i >> 4;
  int m = i & 15;
  float lo = 0.0f, hi = 0.0f;
  if (m < C_DIM) {
    lo = W[(size_t)(2 * kk2) * C_DIM + m];
    hi = W[(size_t)(2 * kk2 + 1) * C_DIM + m];
  }
  Wp2[2 * i] = lo;
  Wp2[2 * i + 1] = hi;
}

// ---------------- K1: degree via scatter-atomics over dst ----------------
__global__ void k_degree(const long long* __restrict__ dst,
                         float* __restrict__ deg, long long E) {
  long long e = (long long)blockIdx.x * blockDim.x + threadIdx.x;
  if (e < E) atomicAdd(&deg[dst[e]], 1.0f);
}

// ---------------- K2: deg -> deg^{-1/2} in place (deg >= 1 always) ----------------
__global__ void k_rsqrt(float* __restrict__ deg, int N) {
  int i = blockIdx.x * blockDim.x + threadIdx.x;
  if (i < N) deg[i] = rsqrtf(deg[i]);
}

// ---------------- K3: x = hiddens @ W via V_WMMA_F32_16X16X4_F32 ----------------
// One wave computes a 16-row tile. Inner step: 1 global_load_b64 (A) +
// 1 ds_load_b64 (B from LDS-staged padded W) + 1 v_wmma. Branch-free.
__global__ void k_gemm_wmma(const float* __restrict__ hiddens,
                            const float* __restrict__ Wp2,
                            float* __restrict__ x, int N) {
  __shared__ v2f Wl[WP2_ELEMS];  // 8 KB
  // Cooperative stage of padded W into LDS (before any early exit).
  for (int t = threadIdx.x; t < WP2_ELEMS; t += blockDim.x)
    Wl[t] = ((const v2f*)Wp2)[t];
  __syncthreads();

  int wave = (int)((blockIdx.x * blockDim.x + threadIdx.x) >> 5);
  int lane = threadIdx.x & 31;
  int tiles = (N + 15) >> 4;
  if (wave >= tiles) return;  // wave-uniform: EXEC stays all-1s for WMMA

  int half = lane >> 4;   // 0: lanes 0-15, 1: lanes 16-31
  int m    = lane & 15;   // A-row / B-column index

  int row = wave * 16 + m;
  int rload = (row < N) ? row : (N - 1);  // clamp tail loads (stores guarded)
  const float* arow = hiddens + (size_t)rload * H_DIM;

  v8f c = {0.f, 0.f, 0.f, 0.f, 0.f, 0.f, 0.f, 0.f};
#pragma unroll
  for (int k = 0; k < H_DIM; k += 4) {
    // A 16x4 layout: VGPR0 = K=kk, VGPR1 = K=kk+1; lanes 16-31 take K+2.
    int kk = k + 2 * half;
    v2f a = *(const v2f*)(arow + kk);
    // B 4x16: same K split; pair (kk, kk+1) pre-interleaved in LDS.
    v2f b = Wl[((k >> 1) + half) * 16 + m];
    // 8 args: (neg_a, A, neg_b, B, c_mod, C, reuse_a, reuse_b)
    c = __builtin_amdgcn_wmma_f32_16x16x4_f32(false, a, false, b, (short)0, c,
                                              false, false);
  }

  // D layout: lane (half,m) VGPR v holds D[M = v + 8*half][N = m]
  if (m < C_DIM) {
    int rb = wave * 16 + 8 * half;
#pragma unroll
    for (int v = 0; v < 8; ++v) {
      int r = rb + v;
      if (r < N) x[(size_t)r * C_DIM + m] = c[v];
    }
  }
}

// ---------------- K4: edge scatter: out[dst] += x[src] * dinv[src]*dinv[dst] ------
__global__ void k_scatter(const long long* __restrict__ src,
                          const long long* __restrict__ dst,
                          const float* __restrict__ x,
                          const float* __restrict__ dinv,
                          float* __restrict__ out, long long E) {
  long long e = (long long)blockIdx.x * blockDim.x + threadIdx.x;
  if (e >= E) return;
  long long s = src[e];
  long long d = dst[e];
  float w = dinv[s] * dinv[d];
  const float4* xs = (const float4*)(x + (size_t)s * C_DIM);
  float4 m0 = xs[0];
  float4 m1 = xs[1];
  float* o = out + (size_t)d * C_DIM;
  atomicAdd(o + 0, m0.x * w);
  atomicAdd(o + 1, m0.y * w);
  atomicAdd(o + 2, m0.z * w);
  atomicAdd(o + 3, m0.w * w);
  atomicAdd(o + 4, m1.x * w);
  atomicAdd(o + 5, m1.y * w);
  atomicAdd(o + 6, m1.z * w);
  atomicAdd(o + 7, m1.w * w);
}

// ---------------- K5: per-node cross-entropy, block reduce, atomic into acc -------
// Self-loop term folded in here: logits = out + x*dinv^2 + b.
__global__ void k_loss(const float* __restrict__ out, const float* __restrict__ x,
                       const float* __restrict__ dinv, const float* __restrict__ b,
                       const long long* __restrict__ y, float* __restrict__ acc,
                       int N) {
  __shared__ float sdata[256];
  int i = blockIdx.x * blockDim.x + threadIdx.x;
  float li = 0.0f;
  if (i < N) {
    float dv = dinv[i];
    float sw = dv * dv;
    float lg[C_DIM];
    float mx = -3.0e38f;
#pragma unroll
    for (int c = 0; c < C_DIM; ++c) {
      lg[c] = out[(size_t)i * C_DIM + c] + x[(size_t)i * C_DIM + c] * sw + b[c];
      mx = fmaxf(mx, lg[c]);
    }
    float se = 0.0f;
#pragma unroll
    for (int c = 0; c < C_DIM; ++c) se += __expf(lg[c] - mx);
    float lse = mx + __logf(se);
    li = lse - lg[(int)y[i]];  // -log_softmax at label
  }
  sdata[threadIdx.x] = li;
  __syncthreads();
  for (int s2 = 128; s2 > 0; s2 >>= 1) {
    if ((int)threadIdx.x < s2) sdata[threadIdx.x] += sdata[threadIdx.x + s2];
    __syncthreads();
  }
  if (threadIdx.x == 0) atomicAdd(acc, sdata[0]);
}

// ---------------- K6: finalize ----------------
__global__ void k_final(const float* __restrict__ acc, float* __restrict__ outp,
                        float invN) {
  if (blockIdx.x == 0 && threadIdx.x == 0) outp[0] = acc[0] * invN;
}

extern "C" void kernel_launch(void* const* d_in, const int* in_sizes, int n_in,
                              void* d_out, int out_size, void* d_ws, size_t ws_size,
                              hipStream_t stream) {
  (void)n_in; (void)out_size; (void)ws_size;
  const float*      hiddens    = (const float*)d_in[0];
  const long long*  edge_index = (const long long*)d_in[1];
  const long long*  y          = (const long long*)d_in[2];
  // d_in[3] = q_edge_index: unused by the reference loss
  const float*      W          = (const float*)d_in[4];
  const float*      b          = (const float*)d_in[5];

  int       N = in_sizes[0] / H_DIM;
  long long E = (long long)in_sizes[1] / 2;
  const long long* src = edge_index;
  const long long* dst = edge_index + E;

  // Workspace layout (floats): x[N*8] | deg/dinv[N] | out[N*8] | acc[1] | Wp2[2048]
  float* ws   = (float*)d_ws;
  float* x    = ws;
  float* deg  = x + (size_t)N * C_DIM;
  float* outb = deg + (size_t)N;
  float* acc  = outb + (size_t)N * C_DIM;
  float* Wp2  = acc + 1;

  const int BT = 256;

  int initN = N * C_DIM;
  k_init<<<(initN + BT - 1) / BT, BT, 0, stream>>>(outb, deg, acc, N);

  k_prepW<<<(WP2_ELEMS + BT - 1) / BT, BT, 0, stream>>>(W, Wp2);

  long long eb = (E + BT - 1) / BT;
  k_degree<<<(int)eb, BT, 0, stream>>>(dst, deg, E);

  k_rsqrt<<<(N + BT - 1) / BT, BT, 0, stream>>>(deg, N);

  int tiles = (N + 15) / 16;                 // one wave per 16-row tile
  int wavesPerBlock = BT / 32;               // 8 waves per 256-thread block
  int gemmBlocks = (tiles + wavesPerBlock - 1) / wavesPerBlock;
  k_gemm_wmma<<<gemmBlocks, BT, 0, stream>>>(hiddens, Wp2, x, N);

  k_scatter<<<(int)eb, BT, 0, stream>>>(src, dst, x, deg, outb, E);

  k_loss<<<(N + BT - 1) / BT, BT, 0, stream>>>(outb, x, deg, b, y, acc, N);

  k_final<<<1, 1, 0, stream>>>(acc, (float*)d_out, 1.0f / (float)N);
}